// IterativeEmbeddingModel_89172110999958
// MI455X (gfx1250) — compile-verified
//
#include <hip/hip_runtime.h>
#include <hip/hip_bf16.h>

#define NN    50000
#define EE    400000
#define PP    64
#define DIMF  192          // 3*P
#define ROWT  3125         // 50000 / 16

typedef __attribute__((ext_vector_type(2))) float v2f;
typedef __attribute__((ext_vector_type(8))) float v8f;

// ---------------------------------------------------------------------------
// Scatter-add aggregation: accum[row[e]] += emb[col[e]]  (DIM=192 f32)
// One thread per (edge, float4-chunk): 48 chunks per edge. Consecutive threads
// walk the 768B contiguous source row -> coalesced loads; atomics are
// GLOBAL_ATOMIC_ADD_F32 (L2-resident: emb + accumulators fit in 192MB L2).
// ---------------------------------------------------------------------------
__global__ void scatter_add_kernel(const float* __restrict__ emb,
                                   const int* __restrict__ rows,
                                   const int* __restrict__ cols,
                                   float* __restrict__ accum,
                                   int nEdges) {
    long long idx = (long long)blockIdx.x * blockDim.x + threadIdx.x;
    int e = (int)(idx / 48);
    int c = (int)(idx % 48);
    if (e >= nEdges) return;
    int s = cols[e];
    int d = rows[e];
    const float4 v = ((const float4*)(emb + (size_t)s * DIMF))[c];
    float* dst = accum + (size_t)d * DIMF + c * 4;
    unsafeAtomicAdd(dst + 0, v.x);
    unsafeAtomicAdd(dst + 1, v.y);
    unsafeAtomicAdd(dst + 2, v.z);
    unsafeAtomicAdd(dst + 3, v.w);
}

// ---------------------------------------------------------------------------
// Fused 3-way GEMM: out[:, m*64 : (m+1)*64] = src_m @ theta_m, m = blockIdx.y
// Block = 128 threads = 4 waves; wave ct owns columns [ct*16, ct*16+16) of a
// 16-row tile. K=192 swept in 48 steps of V_WMMA_F32_16X16X4_F32 (exact f32).
// ---------------------------------------------------------------------------
__global__ void __launch_bounds__(128)
gemm3_wmma_kernel(const float* __restrict__ src0,   // emb
                  const float* __restrict__ src1,   // nbr_sum
                  const float* __restrict__ src2,   // anti_sum
                  const float* __restrict__ th1,
                  const float* __restrict__ th2,
                  const float* __restrict__ th3,
                  float* __restrict__ out) {
    const int mat = blockIdx.y;                       // 0..2 (uniform per block)
    const float* __restrict__ src = (mat == 0) ? src0 : ((mat == 1) ? src1 : src2);
    const float* __restrict__ th  = (mat == 0) ? th1  : ((mat == 1) ? th2  : th3);

    const int row0 = blockIdx.x * 16;                 // 16-row output tile
    const int ct   = threadIdx.x >> 5;                // wave id -> column tile (0..3)
    const int lane = threadIdx.x & 31;
    const int half = lane >> 4;                       // 0: lanes 0-15, 1: lanes 16-31
    const int mn   = lane & 15;                       // M index for A, N index for B/C/D

    const int ncol = ct * 16 + mn;                    // column within this theta (0..63)

    v8f acc = {};

    // A fragment base: row (row0+mn), element (k0 + 2*half) -> aligned float2
    const float* arow = src + (size_t)(row0 + mn) * DIMF + 2 * half;

    #pragma unroll 4
    for (int k0 = 0; k0 < DIMF; k0 += 4) {
        v2f a = *(const v2f*)(arow + k0);
        v2f b;
        b.x = th[(size_t)(k0 + 2 * half)     * PP + ncol];
        b.y = th[(size_t)(k0 + 2 * half + 1) * PP + ncol];
        acc = __builtin_amdgcn_wmma_f32_16x16x4_f32(
            /*neg_a=*/false, a, /*neg_b=*/false, b,
            /*c_mod=*/(short)0, acc, /*reuse_a=*/false, /*reuse_b=*/false);
    }

    // D layout: VGPR r, lanes 0-15 -> M=r ; lanes 16-31 -> M=r+8 ; N = lane&15
    float* obase = out + (size_t)mat * PP + ncol;
    #pragma unroll
    for (int r = 0; r < 8; ++r) {
        obase[(size_t)(row0 + r + 8 * half) * DIMF] = acc[r];
    }
}

// ---------------------------------------------------------------------------
// Launch: two iterations. Workspace: [emb_iter1 | nbr | anti]  (3 * 38.4 MB)
// ---------------------------------------------------------------------------
extern "C" void kernel_launch(void* const* d_in, const int* in_sizes, int n_in,
                              void* d_out, int out_size, void* d_ws, size_t ws_size,
                              hipStream_t stream) {
    const float* emb0 = (const float*)d_in[0];
    const float* th1  = (const float*)d_in[1];
    const float* th2  = (const float*)d_in[2];
    const float* th3  = (const float*)d_in[3];
    const int*   ei   = (const int*)d_in[4];   // (2, E): rows then cols
    const int*   aei  = (const int*)d_in[5];
    const int    E    = in_sizes[4] / 2;

    const size_t embElems = (size_t)NN * DIMF;
    float* ws_emb = (float*)d_ws;              // iteration-1 output
    float* nbr    = ws_emb + embElems;
    float* anti   = nbr + embElems;
    const size_t embBytes = embElems * sizeof(float);

    const int scatterThreads = 256;
    const long long scatterWork = (long long)E * 48;
    const int scatterBlocks = (int)((scatterWork + scatterThreads - 1) / scatterThreads);

    dim3 gemmGrid(ROWT, 3);
    dim3 gemmBlock(128);

    // ---- iteration 1: input = d_in embeddings, output = ws_emb ----
    hipMemsetAsync(nbr,  0, embBytes, stream);
    hipMemsetAsync(anti, 0, embBytes, stream);
    scatter_add_kernel<<<scatterBlocks, scatterThreads, 0, stream>>>(
        emb0, ei,  ei  + E, nbr,  E);
    scatter_add_kernel<<<scatterBlocks, scatterThreads, 0, stream>>>(
        emb0, aei, aei + E, anti, E);
    gemm3_wmma_kernel<<<gemmGrid, gemmBlock, 0, stream>>>(
        emb0, nbr, anti, th1, th2, th3, ws_emb);

    // ---- iteration 2: input = ws_emb, output = d_out ----
    hipMemsetAsync(nbr,  0, embBytes, stream);
    hipMemsetAsync(anti, 0, embBytes, stream);
    scatter_add_kernel<<<scatterBlocks, scatterThreads, 0, stream>>>(
        ws_emb, ei,  ei  + E, nbr,  E);
    scatter_add_kernel<<<scatterBlocks, scatterThreads, 0, stream>>>(
        ws_emb, aei, aei + E, anti, E);
    gemm3_wmma_kernel<<<gemmGrid, gemmBlock, 0, stream>>>(
        ws_emb, nbr, anti, th1, th2, th3, (float*)d_out);
}